// Sampler_85607288144299
// MI455X (gfx1250) — compile-verified
//
#include <hip/hip_runtime.h>
#include <stdint.h>

#define VOCAB_   128000
#define THREADS_ 640                   // 20 wave32s; 32000 float4 / 640 = 50 exact
#define NVEC_    (VOCAB_ / 4)          // 32000 float4 per row
#define ITERS_   (NVEC_ / THREADS_)    // 50
#define NWAVES_  (THREADS_ / 32)       // 20
#define NBUF_    4                     // 4-deep async ring (80 KB LDS / block)

// gfx1250 async global->LDS path (GLOBAL_LOAD_ASYNC_TO_LDS_B128, ASYNCcnt)
#if defined(__has_builtin)
#  if __has_builtin(__builtin_amdgcn_global_load_async_to_lds_b128)
#    define USE_ASYNC_LDS 1
#  endif
#endif

#if defined(USE_ASYNC_LDS)
typedef int v4i_ __attribute__((vector_size(4 * sizeof(int))));
typedef __attribute__((address_space(1))) v4i_* gv4i_p;   // global (as1)
typedef __attribute__((address_space(3))) v4i_* lv4i_p;   // LDS (as3)
#define ASYNC_CP_B128(gp, lp)                                                  \
  __builtin_amdgcn_global_load_async_to_lds_b128((gv4i_p)(gp), (lv4i_p)(lp),   \
                                                 0, 0)
#endif

#define INV_LN2_ 1.4426950408889634f

__device__ __forceinline__ void amax_upd(float v, int i, float& bv, int& bi) {
  if (v > bv) { bv = v; bi = i; }   // strict > keeps earliest index (jnp.argmax)
}

__device__ __forceinline__ void wave_amax(float& bv, int& bi) {
#pragma unroll
  for (int off = 16; off > 0; off >>= 1) {   // wave32
    float ov = __shfl_xor(bv, off, 32);
    int   oi = __shfl_xor(bi, off, 32);
    if (ov > bv || (ov == bv && oi < bi)) { bv = ov; bi = oi; }
  }
}

__global__ __launch_bounds__(THREADS_) void Sampler_85607288144299_kernel(
    const float* __restrict__ logits, const float* __restrict__ temps,
    const float* __restrict__ noise, int* __restrict__ out) {
#if defined(USE_ASYNC_LDS)
  __shared__ float4 s_lg[NBUF_][THREADS_];   // 40 KB
  __shared__ float4 s_nz[NBUF_][THREADS_];   // 40 KB
#endif
  __shared__ float s_v[NWAVES_];
  __shared__ int   s_i[NWAVES_];

  const int row = blockIdx.x;
  const int tid = threadIdx.x;
  const float t = temps[row];

  const float4* __restrict__ lg =
      reinterpret_cast<const float4*>(logits) + (size_t)row * NVEC_;
  const float4* __restrict__ nz =
      reinterpret_cast<const float4*>(noise) + (size_t)row * NVEC_;

  float bv = -3.402823466e38f;
  int   bi = 0x7fffffff;

  if (t > 0.0f) {
    // argmax(probs/noise) == argmax(x/T - ln(max(n,1e-10)))
    //                     == argmax(x*(invT/ln2) - log2(max(n,1e-10)))  (monotone)
    const float invT2 = (1.0f / t) * INV_LN2_;
#if defined(USE_ASYNC_LDS)
    // ---- 4-deep ring of async global->LDS tiles (CDNA5 ASYNCcnt path) ----
#pragma unroll
    for (int p = 0; p < NBUF_; ++p) {
      ASYNC_CP_B128(lg + p * THREADS_ + tid, &s_lg[p][tid]);
      ASYNC_CP_B128(nz + p * THREADS_ + tid, &s_nz[p][tid]);
    }
    for (int it = 0; it < ITERS_; ++it) {
      const int ahead = ITERS_ - 1 - it;   // younger tiles still in flight
      if (ahead >= NBUF_ - 1) {
        asm volatile("s_wait_asynccnt 6" ::: "memory");
      } else if (ahead == 2) {
        asm volatile("s_wait_asynccnt 4" ::: "memory");
      } else if (ahead == 1) {
        asm volatile("s_wait_asynccnt 2" ::: "memory");
      } else {
        asm volatile("s_wait_asynccnt 0" ::: "memory");
      }
      const int buf = it & (NBUF_ - 1);
      float4 x = s_lg[buf][tid];
      float4 n = s_nz[buf][tid];
      const int base = (it * THREADS_ + tid) * 4;
      amax_upd(x.x * invT2 - __log2f(fmaxf(n.x, 1e-10f)), base + 0, bv, bi);
      amax_upd(x.y * invT2 - __log2f(fmaxf(n.y, 1e-10f)), base + 1, bv, bi);
      amax_upd(x.z * invT2 - __log2f(fmaxf(n.z, 1e-10f)), base + 2, bv, bi);
      amax_upd(x.w * invT2 - __log2f(fmaxf(n.w, 1e-10f)), base + 3, bv, bi);
      if (it + NBUF_ < ITERS_) {
        // DS reads are unordered vs async LDS writes: drain before WAR reuse
        asm volatile("s_wait_dscnt 0" ::: "memory");
        ASYNC_CP_B128(lg + (it + NBUF_) * THREADS_ + tid, &s_lg[buf][tid]);
        ASYNC_CP_B128(nz + (it + NBUF_) * THREADS_ + tid, &s_nz[buf][tid]);
      }
    }
#else
    for (int it = 0; it < ITERS_; ++it) {
      const int v = it * THREADS_ + tid;
      if (it + 4 < ITERS_) {
        __builtin_prefetch(lg + v + 4 * THREADS_, 0, 0);  // global_prefetch_b8
        __builtin_prefetch(nz + v + 4 * THREADS_, 0, 0);
      }
      float4 x = lg[v];
      float4 n = nz[v];
      const int base = v * 4;
      amax_upd(x.x * invT2 - __log2f(fmaxf(n.x, 1e-10f)), base + 0, bv, bi);
      amax_upd(x.y * invT2 - __log2f(fmaxf(n.y, 1e-10f)), base + 1, bv, bi);
      amax_upd(x.z * invT2 - __log2f(fmaxf(n.z, 1e-10f)), base + 2, bv, bi);
      amax_upd(x.w * invT2 - __log2f(fmaxf(n.w, 1e-10f)), base + 3, bv, bi);
    }
#endif
  } else {
    // greedy: argmax over raw logits only (half the bytes for these rows)
    for (int it = 0; it < ITERS_; ++it) {
      const int v = it * THREADS_ + tid;
      if (it + 4 < ITERS_) {
        __builtin_prefetch(lg + v + 4 * THREADS_, 0, 0);
      }
      float4 x = lg[v];
      const int base = v * 4;
      amax_upd(x.x, base + 0, bv, bi);
      amax_upd(x.y, base + 1, bv, bi);
      amax_upd(x.z, base + 2, bv, bi);
      amax_upd(x.w, base + 3, bv, bi);
    }
  }

  // ---- cross-lane then cross-wave argmax reduction (wave32) ----
  wave_amax(bv, bi);
  const int wave = tid >> 5, lane = tid & 31;
  if (lane == 0) { s_v[wave] = bv; s_i[wave] = bi; }
  __syncthreads();
  if (wave == 0) {
    float v = (lane < NWAVES_) ? s_v[lane] : -3.402823466e38f;
    int   i = (lane < NWAVES_) ? s_i[lane] : 0x7fffffff;
    wave_amax(v, i);
    if (lane == 0) out[row] = i;
  }
}

extern "C" void kernel_launch(void* const* d_in, const int* in_sizes, int n_in,
                              void* d_out, int out_size, void* d_ws, size_t ws_size,
                              hipStream_t stream) {
  const float* logits = (const float*)d_in[0];
  const float* temps  = (const float*)d_in[1];
  const float* noise  = (const float*)d_in[2];
  const int batch = in_sizes[1];  // 256 rows (one block per row)
  Sampler_85607288144299_kernel<<<batch, THREADS_, 0, stream>>>(
      logits, temps, noise, (int*)d_out);
}